// GSB_Mamba_87368224735809
// MI455X (gfx1250) — compile-verified
//
#include <hip/hip_runtime.h>
#include <hip/hip_bf16.h>

// ---------------------------------------------------------------------------
// Vision-Mamba block for MI455X (gfx1250), fp32 end-to-end.
// - Dense layers: V_WMMA_F32_16X16X4_F32 (wave32 WMMA, fp32-exact vs reference)
// - Big GEMMs stage the B tile via the Tensor Data Mover (tensor_load_to_lds,
//   TENSORcnt double-buffered); A tile via cooperative loads. No bounds guards.
// - Small/ragged GEMMs (x_proj M=40, dt_proj K=8) use a guarded WMMA kernel.
// - Selective scan: registers hold h[16]; B/C staged in LDS; prefetch of
//   streaming rows via global_prefetch_b8.
// Shapes: B=4, CIN=192, COUT=128, H=W=64 (L=4096), ED=256, N=16, R=8, K=4
// ---------------------------------------------------------------------------

#define BB   4
#define CIN  192
#define CO   128
#define LN   4096      // H*W
#define ED   256
#define NS   16
#define RR   8
#define KK   4

typedef __attribute__((ext_vector_type(2))) float        v2f;
typedef __attribute__((ext_vector_type(8))) float        v8f;
typedef __attribute__((ext_vector_type(4))) unsigned int v4u;
typedef __attribute__((ext_vector_type(4))) int          v4i;
typedef __attribute__((ext_vector_type(8))) int          v8i;

#if __has_builtin(__builtin_amdgcn_tensor_load_to_lds) && \
    __has_builtin(__builtin_amdgcn_s_wait_tensorcnt)
#define HAVE_TDM 1
#else
#define HAVE_TDM 0
#endif

#if HAVE_TDM
// ---------------------------------------------------------------------------
// TDM: DMA a 16(k) x 64(n) f32 tile from a row-major (K x LN) matrix into LDS.
// D# built per CDNA5 ISA 8.3/8.4: group0 = {count=1, lds_addr, global_addr,
// type=2}; group1 = {data_size=4B, tensor_dim0=LN, tensor_dim1=K,
// tile_dim0=64, tile_dim1=16, tensor_dim0_stride=LN}. 2D -> groups 2/3 zero.
// ---------------------------------------------------------------------------
__device__ __forceinline__ void tdm_load_b_tile(const float* gtile,
                                                unsigned int lds_off, int K)
{
    unsigned long long ga = (unsigned long long)(uintptr_t)gtile;
    v4u g0 = { 1u,                                             // count=1, user D#
               lds_off,                                        // lds_addr
               (unsigned int)(ga & 0xFFFFFFFFu),               // global_addr lo
               (unsigned int)((ga >> 32) & 0x01FFFFFFu)        // global_addr hi
                   | (2u << 30) };                             // type=2 (image)
    v8i g1 = { (int)(2u << 16),                // data_size=2 (4 bytes)
               (int)((unsigned)LN << 16),      // tensor_dim0[15:0] in [31:16]
               (int)((unsigned)K  << 16),      // dim0 hi=0 | tensor_dim1 lo
               (int)(64u << 16),               // dim1 hi=0 | tile_dim0=64
               16,                             // tile_dim1=16 | tile_dim2=0
               LN,                             // tensor_dim0_stride lo32
               0, 0 };                         // stride hi | dim1_stride=0
    v4i z4 = {0, 0, 0, 0};
#if defined(__clang_major__) && (__clang_major__ >= 23)
    v8i z8 = {0, 0, 0, 0, 0, 0, 0, 0};
    __builtin_amdgcn_tensor_load_to_lds(g0, g1, z4, z4, z8, 0);
#else
    __builtin_amdgcn_tensor_load_to_lds(g0, g1, z4, z4, 0);
#endif
}
#endif

// ---------------------------------------------------------------------------
// Fast WMMA GEMM (no guards): M % 32 == 0, K % 16 == 0.
//   Yb[M x LN] = W[M x K] * Xb[K x LN]  (+ epilogue)
// B tile TDM-staged + double-buffered; A tile cooperative.
// Block 256 = 8 waves; wave tile 16x16; block tile 32(M) x 64(N).
// ---------------------------------------------------------------------------
template <int EPI>
__global__ __launch_bounds__(256) void wmma_gemm_fast(
    const float* __restrict__ W, const float* __restrict__ X,
    float* __restrict__ Y, int M, int K,
    const float* __restrict__ bias, const float* __restrict__ extra)
{
    const int b  = blockIdx.z;
    const int n0 = blockIdx.x * 64;
    const int m0 = blockIdx.y * 32;
    const float* Xb = X + (size_t)b * K * LN;
    float*       Yb = Y + (size_t)b * M * LN;

    __shared__ float As[16][33];        // [k][m], padded
    __shared__ float Bs[2][16][64];     // TDM writes row-major tiles

    const int tid  = threadIdx.x;
    const int lane = tid & 31;
    const int wave = tid >> 5;
    const int wm   = wave >> 2;
    const int wn   = wave & 3;
    const int l15  = lane & 15;
    const int half = lane >> 4;

    const int nIter = K >> 4;

#if HAVE_TDM
    const unsigned int lds_b[2] = {
        (unsigned int)(uintptr_t)&Bs[0][0][0],
        (unsigned int)(uintptr_t)&Bs[1][0][0] };
    if (tid == 0)                                   // TDM ignores EXEC; one wave issues
        tdm_load_b_tile(Xb + n0, lds_b[0], K);
#endif

    v8f acc = {0.f, 0.f, 0.f, 0.f, 0.f, 0.f, 0.f, 0.f};

    for (int it = 0; it < nIter; ++it) {
        const int k0  = it << 4;
        const int cur = it & 1;

        // --- cooperative A tile: 32 rows x 16 k, 2 floats/thread, no guards ---
        {
            const int r = tid >> 3, c = (tid & 7) << 1;
            const float* wp = W + (size_t)(m0 + r) * K + k0 + c;
            const float a0 = wp[0], a1 = wp[1];
            As[c][r]     = a0;
            As[c + 1][r] = a1;
        }

#if HAVE_TDM
        if (tid == 0) {
            if (it + 1 < nIter) {
                tdm_load_b_tile(Xb + (size_t)(k0 + 16) * LN + n0,
                                lds_b[cur ^ 1], K);
                __builtin_amdgcn_s_wait_tensorcnt(1);  // current tile landed
            } else {
                __builtin_amdgcn_s_wait_tensorcnt(0);  // last tile landed
            }
        }
#else
        {   // fallback cooperative B tile load
            const int r = tid >> 6, c = tid & 63;
            #pragma unroll
            for (int rr2 = 0; rr2 < 16; rr2 += 4)
                Bs[cur][r + rr2][c] = Xb[(size_t)(k0 + r + rr2) * LN + n0 + c];
        }
#endif
        __syncthreads();

        #pragma unroll
        for (int ks = 0; ks < 16; ks += 4) {
            const int ka = ks + half * 2;
            v2f a, bv;
            a[0]  = As[ka    ][wm * 16 + l15];
            a[1]  = As[ka + 1][wm * 16 + l15];
            bv[0] = Bs[cur][ka    ][wn * 16 + l15];
            bv[1] = Bs[cur][ka + 1][wn * 16 + l15];
            acc = __builtin_amdgcn_wmma_f32_16x16x4_f32(
                      false, a, false, bv, (short)0, acc, false, false);
        }
        __syncthreads();    // all waves done with As / Bs[cur] before reuse
    }

    const int col = n0 + wn * 16 + l15;
    #pragma unroll
    for (int r = 0; r < 8; ++r) {
        const int row = m0 + wm * 16 + r + (half ? 8 : 0);
        float v = acc[r];
        if (EPI == 1)      v += bias[row];
        else if (EPI == 2) v += bias[row] + extra[(size_t)row * LN + col];
        else if (EPI == 4) v += extra[(size_t)b * M * LN + (size_t)row * LN + col];
        Yb[(size_t)row * LN + col] = v;
    }
}

// ---------------------------------------------------------------------------
// Guarded WMMA GEMM for ragged shapes (x_proj M=40, dt_proj K=8).
// Epilogues: 0 none | 3 softplus(+bias)
// ---------------------------------------------------------------------------
template <int EPI>
__global__ __launch_bounds__(256) void wmma_gemm_guarded(
    const float* __restrict__ W, const float* __restrict__ X,
    float* __restrict__ Y, int M, int K, int KbX,
    const float* __restrict__ bias)
{
    const int b  = blockIdx.z;
    const int n0 = blockIdx.x * 64;
    const int m0 = blockIdx.y * 32;
    const float* Xb = X + (size_t)b * KbX * LN;
    float*       Yb = Y + (size_t)b * M   * LN;

    __shared__ float As[16][33];
    __shared__ float Bs[16][65];

    const int tid  = threadIdx.x;
    const int lane = tid & 31;
    const int wave = tid >> 5;
    const int wm   = wave >> 2;
    const int wn   = wave & 3;
    const int l15  = lane & 15;
    const int half = lane >> 4;

    v8f acc = {0.f, 0.f, 0.f, 0.f, 0.f, 0.f, 0.f, 0.f};

    for (int k0 = 0; k0 < K; k0 += 16) {
        {
            int r = tid >> 4, c = tid & 15;
            #pragma unroll
            for (int rr2 = 0; rr2 < 32; rr2 += 16) {
                int row = m0 + r + rr2, colk = k0 + c;
                As[c][r + rr2] = (row < M && colk < K) ? W[(size_t)row * K + colk] : 0.f;
            }
        }
        {
            int r = tid >> 6, c = tid & 63;
            #pragma unroll
            for (int rr2 = 0; rr2 < 16; rr2 += 4) {
                int row = k0 + r + rr2;
                Bs[r + rr2][c] = (row < K) ? Xb[(size_t)row * LN + n0 + c] : 0.f;
            }
        }
        __syncthreads();
        #pragma unroll
        for (int ks = 0; ks < 16; ks += 4) {
            const int ka = ks + half * 2;
            v2f a, bv;
            a[0]  = As[ka    ][wm * 16 + l15];
            a[1]  = As[ka + 1][wm * 16 + l15];
            bv[0] = Bs[ka    ][wn * 16 + l15];
            bv[1] = Bs[ka + 1][wn * 16 + l15];
            acc = __builtin_amdgcn_wmma_f32_16x16x4_f32(
                      false, a, false, bv, (short)0, acc, false, false);
        }
        __syncthreads();
    }

    const int col = n0 + wn * 16 + l15;
    #pragma unroll
    for (int r = 0; r < 8; ++r) {
        int row = m0 + wm * 16 + r + (half ? 8 : 0);
        if (row < M) {
            float v = acc[r];
            if (EPI == 3) {
                v += bias[row];
                v = (v > 20.f) ? v : log1pf(__expf(v));       // softplus
            }
            Yb[(size_t)row * LN + col] = v;
        }
    }
}

// ---------------------------------------------------------------------------
// Sin/cos positional encoding
// ---------------------------------------------------------------------------
__global__ void pe_kernel(float* __restrict__ pe)
{
    int i = blockIdx.x * blockDim.x + threadIdx.x;
    if (i >= CO * LN) return;
    int p = i & (LN - 1), c = i >> 12;
    int gy = p >> 6, gx = p & 63;
    int j = c & 31, sel = c >> 5;
    float omega = __powf(10000.f, -(float)j / 32.f);
    float arg = (sel < 2 ? (float)gx : (float)gy) * omega;
    pe[i] = ((sel & 1) == 0) ? __sinf(arg) : __cosf(arg);
}

// ---------------------------------------------------------------------------
// Depthwise causal conv1d (K=4, left pad 3) + bias + SiLU
// ---------------------------------------------------------------------------
__global__ void conv_silu_kernel(const float* __restrict__ xz,
                                 const float* __restrict__ conv_w,
                                 const float* __restrict__ conv_b,
                                 float* __restrict__ u_act)
{
    int i = blockIdx.x * blockDim.x + threadIdx.x;
    if (i >= BB * ED * LN) return;
    int l = i & (LN - 1);
    int be = i >> 12;
    int e = be & (ED - 1);
    int b = be >> 8;
    const float* row = xz + ((size_t)b * 2 * ED + e) * LN;
    float s = conv_b[e];
    #pragma unroll
    for (int k = 0; k < KK; ++k) {
        int ls = l - (KK - 1) + k;
        if (ls >= 0) s = fmaf(conv_w[e * KK + k], row[ls], s);
    }
    u_act[i] = s / (1.f + __expf(-s));   // silu
}

// ---------------------------------------------------------------------------
// Selective scan. grid = (B, ED/64), block 64; thread e keeps h[16] + A[16]
// in registers; B/C rows of x_dbl staged in LDS per 64-step chunk.
// Next-chunk streaming rows prefetched (global_prefetch_b8).
// ---------------------------------------------------------------------------
__global__ __launch_bounds__(64) void scan_kernel(
    const float* __restrict__ dt_act,   // (B, ED, LN) softplus(dt) already
    const float* __restrict__ u_act,    // (B, ED, LN)
    const float* __restrict__ x_dbl,    // (B, 40, LN): dt 0..7, B 8..23, C 24..39
    const float* __restrict__ xz,       // z = rows [ED, 2*ED)
    const float* __restrict__ A_log,    // (ED, NS)
    const float* __restrict__ D_param,  // (ED)
    float* __restrict__ y_act)          // (B, ED, LN)
{
    const int b = blockIdx.x;
    const int e = blockIdx.y * 64 + threadIdx.x;

    __shared__ float Bc[64][NS];
    __shared__ float Cc[64][NS];

    float A[NS], h[NS];
    #pragma unroll
    for (int n = 0; n < NS; ++n) { A[n] = -__expf(A_log[e * NS + n]); h[n] = 0.f; }
    const float Dv = D_param[e];

    const float* dtp = dt_act + ((size_t)b * ED + e) * LN;
    const float* up  = u_act  + ((size_t)b * ED + e) * LN;
    const float* zp  = xz     + ((size_t)b * 2 * ED + ED + e) * LN;
    float*       yp  = y_act  + ((size_t)b * ED + e) * LN;
    const float* xb  = x_dbl  + (size_t)b * (RR + 2 * NS) * LN;

    for (int l0 = 0; l0 < LN; l0 += 64) {
        if (l0 + 64 < LN) {        // prefetch next chunk of the streaming rows
            __builtin_prefetch(dtp + l0 + 64, 0, 1);
            __builtin_prefetch(up  + l0 + 64, 0, 1);
            __builtin_prefetch(zp  + l0 + 64, 0, 1);
        }
        __syncthreads();
        #pragma unroll
        for (int it = 0; it < 16; ++it) {
            int idx = threadIdx.x + it * 64;   // 0..1023
            int n = idx >> 6, t = idx & 63;
            Bc[t][n] = xb[(size_t)(RR + n)      * LN + l0 + t];
            Cc[t][n] = xb[(size_t)(RR + NS + n) * LN + l0 + t];
        }
        __syncthreads();

        for (int t = 0; t < 64; ++t) {
            const int l = l0 + t;
            const float dtv = dtp[l];
            const float uv  = up[l];
            const float du  = dtv * uv;
            float y = 0.f;
            #pragma unroll
            for (int n = 0; n < NS; ++n) {
                float a = __expf(dtv * A[n]);
                h[n] = fmaf(a, h[n], du * Bc[t][n]);
                y    = fmaf(h[n], Cc[t][n], y);
            }
            const float zv = zp[l];
            float yv = fmaf(uv, Dv, y);
            yv *= zv / (1.f + __expf(-zv));    // * silu(z)
            yp[l] = yv;
        }
    }
}

// ---------------------------------------------------------------------------
// Launch: inputs in setup_inputs() order.
// ---------------------------------------------------------------------------
extern "C" void kernel_launch(void* const* d_in, const int* in_sizes, int n_in,
                              void* d_out, int out_size, void* d_ws, size_t ws_size,
                              hipStream_t stream)
{
    const float* x          = (const float*)d_in[0];
    const float* proj_w     = (const float*)d_in[1];
    const float* proj_b     = (const float*)d_in[2];
    const float* lp_w       = (const float*)d_in[3];
    const float* lp_b       = (const float*)d_in[4];
    const float* in_proj_w  = (const float*)d_in[5];
    const float* conv_w     = (const float*)d_in[6];
    const float* conv_b     = (const float*)d_in[7];
    const float* x_proj_w   = (const float*)d_in[8];
    const float* dt_proj_w  = (const float*)d_in[9];
    const float* dt_proj_b  = (const float*)d_in[10];
    const float* A_log      = (const float*)d_in[11];
    const float* D_param    = (const float*)d_in[12];
    const float* out_proj_w = (const float*)d_in[13];

    float* ws = (float*)d_ws;
    size_t off = 0;
    float* pe     = ws + off; off += (size_t)CO * LN;
    float* h_buf  = ws + off; off += (size_t)BB * CO * LN;
    float* seq    = ws + off; off += (size_t)BB * CO * LN;
    float* xz     = ws + off; off += (size_t)BB * 2 * ED * LN;
    float* u_act  = ws + off; off += (size_t)BB * ED * LN;
    float* x_dbl  = ws + off; off += (size_t)BB * (RR + 2 * NS) * LN;
    float* dt_act = ws + off; off += (size_t)BB * ED * LN;
    float* y_act  = ws + off; off += (size_t)BB * ED * LN;
    (void)ws_size; (void)in_sizes; (void)n_in; (void)out_size;

    float* out = (float*)d_out;

    // 1) positional encoding
    pe_kernel<<<(CO * LN + 255) / 256, 256, 0, stream>>>(pe);

    // 2) h = proj_w @ x + proj_b + pe           (M=128, K=192)
    wmma_gemm_fast<2><<<dim3(LN/64, CO/32, BB), 256, 0, stream>>>(
        proj_w, x, h_buf, CO, CIN, proj_b, pe);

    // 3) seq = lp_w @ h + lp_b                  (M=128, K=128)
    wmma_gemm_fast<1><<<dim3(LN/64, CO/32, BB), 256, 0, stream>>>(
        lp_w, h_buf, seq, CO, CO, lp_b, nullptr);

    // 4) xz = in_proj_w @ seq                   (M=512, K=128)
    wmma_gemm_fast<0><<<dim3(LN/64, (2*ED)/32, BB), 256, 0, stream>>>(
        in_proj_w, seq, xz, 2*ED, CO, nullptr, nullptr);

    // 5) depthwise causal conv + bias + silu -> u_act
    conv_silu_kernel<<<(BB * ED * LN + 255) / 256, 256, 0, stream>>>(
        xz, conv_w, conv_b, u_act);

    // 6) x_dbl = x_proj_w @ u_act               (M=40 ragged, K=256)
    wmma_gemm_guarded<0><<<dim3(LN/64, 2, BB), 256, 0, stream>>>(
        x_proj_w, u_act, x_dbl, RR + 2*NS, ED, ED, nullptr);

    // 7) dt_act = softplus(dt_proj_w @ x_dbl[0:8] + dt_proj_b)  (M=256, K=8)
    wmma_gemm_guarded<3><<<dim3(LN/64, ED/32, BB), 256, 0, stream>>>(
        dt_proj_w, x_dbl, dt_act, ED, RR, RR + 2*NS, dt_proj_b);

    // 8) selective scan -> y_act
    scan_kernel<<<dim3(BB, ED/64), 64, 0, stream>>>(
        dt_act, u_act, x_dbl, xz, A_log, D_param, y_act);

    // 9) out = out_proj_w @ y_act + h           (M=128, K=256)
    wmma_gemm_fast<4><<<dim3(LN/64, CO/32, BB), 256, 0, stream>>>(
        out_proj_w, y_act, out, CO, ED, nullptr, h_buf);
}